// SelfAttention_44890998178013
// MI455X (gfx1250) — compile-verified
//
#include <hip/hip_runtime.h>

typedef __attribute__((ext_vector_type(2))) float v2f;
typedef __attribute__((ext_vector_type(8))) float v8f;

#define NB 4               // batch items per workgroup (N=131072 divisible by 4)
#define ROWS_P (NB * 36)   // 144 projection rows per block
#define XPITCH 12          // zero-padded K pitch (9 -> 12 = 3 WMMA K-steps)
#define QKV_PITCH 16

static __device__ __forceinline__ int imin(int a, int b) { return a < b ? a : b; }

// Self-attention, N independent tiny heads. Memory-bound (~680MB @ 23.3TB/s ~29us),
// so fp32 throughout; dense GEMMs use v_wmma_f32_16x16x4_f32 with zero-padded LDS
// operands (branchless fragments); the per-(n,h) 9x9 softmax core stays on VALU
// (its K/V operands differ per n -> not WMMA-batchable). All global traffic is
// coalesced float4 (b128) through LDS staging buffers.
__global__ __launch_bounds__(256) void
attn_kernel(const float* __restrict__ values,
            const float* __restrict__ keys,
            const float* __restrict__ query,
            const float* __restrict__ Wv,
            const float* __restrict__ Wk,
            const float* __restrict__ Wq,
            const float* __restrict__ Wo,
            const float* __restrict__ bo,
            float* __restrict__ out,
            int n_total)
{
    __shared__ float sQ[ROWS_P * QKV_PITCH];      // 2304 floats
    __shared__ float sK[ROWS_P * QKV_PITCH];
    __shared__ float sV[ROWS_P * QKV_PITCH];
    // Scratch region, reused:
    //   phase 0/1: sX[3][144*12]   (padded inputs)           5184 floats
    //   phase 2/3: sO[1296] @ +0, sY[1296] @ +1296
    __shared__ float sScratch[3 * ROWS_P * XPITCH];
    __shared__ float sWp[3 * 16 * XPITCH];        // padded Wv/Wk/Wq, pitch 12
    __shared__ float sWo[36 * 36];
    __shared__ float sbo[36];

    float* const sO = sScratch;
    float* const sY = sScratch + NB * 9 * 36;

    const int tid  = threadIdx.x;
    const int wave = tid >> 5;
    const int lane = tid & 31;
    const int half = lane >> 4;   // K-pair selector per f32 WMMA A/B VGPR layout
    const int l16  = lane & 15;
    const int n0   = blockIdx.x * NB;
    const int nv   = imin(NB, n_total - n0);      // valid n's this block

    // ---- Phase 0: stage weights (padded) and inputs (coalesced b128 -> padded LDS)
    for (int i = tid; i < 16 * XPITCH; i += 256) {
        const int j = i / XPITCH, k = i % XPITCH;
        const bool v = (j < 9) & (k < 9);
        sWp[0 * 192 + i] = v ? Wv[j * 9 + k] : 0.f;
        sWp[1 * 192 + i] = v ? Wk[j * 9 + k] : 0.f;
        sWp[2 * 192 + i] = v ? Wq[j * 9 + k] : 0.f;
    }
    for (int i = tid; i < 324; i += 256)
        ((float4*)sWo)[i] = ((const float4*)Wo)[i];
    if (tid < 36) sbo[tid] = bo[tid];

    {
        const int nf4 = nv * 81;                  // float4's per input matrix
        const size_t base = (size_t)n0 * 324;
        const float4* sv0 = (const float4*)(values + base);
        const float4* sv1 = (const float4*)(keys   + base);
        const float4* sv2 = (const float4*)(query  + base);
        for (int i = tid; i < nf4; i += 256) {
            const float4 x0 = sv0[i], x1 = sv1[i], x2 = sv2[i];
            const int e = 4 * i;
            float xs0[4] = {x0.x, x0.y, x0.z, x0.w};
            float xs1[4] = {x1.x, x1.y, x1.z, x1.w};
            float xs2[4] = {x2.x, x2.y, x2.z, x2.w};
#pragma unroll
            for (int c = 0; c < 4; ++c) {
                const int ee = e + c;
                const int r = ee / 9, k = ee % 9;   // scatter to pitch-12, pad stays 0
                const int d = r * XPITCH + k;
                sScratch[0 * 1728 + d] = xs0[c];
                sScratch[1 * 1728 + d] = xs1[c];
                sScratch[2 * 1728 + d] = xs2[c];
            }
        }
        // zero the pad columns (k = 9..11) so WMMA K-padding reads zeros
        for (int r = tid; r < ROWS_P; r += 256) {
#pragma unroll
            for (int m = 0; m < 3; ++m) {
                sScratch[m * 1728 + r * XPITCH + 9]  = 0.f;
                sScratch[m * 1728 + r * XPITCH + 10] = 0.f;
                sScratch[m * 1728 + r * XPITCH + 11] = 0.f;
            }
        }
    }
    __syncthreads();

    // ---- Phase 1: Q/K/V projections. Y(144x9) = X(144x9) @ W^T via WMMA f32 16x16x4.
    // 27 row-tiles (3 matrices x 9 tiles) x 3 K-steps; fully branchless fragments.
    for (int t = wave; t < 27; t += 8) {
        const int mat = t / 9, rt = t % 9;        // wave-uniform
        const float* Xs = sScratch + mat * 1728;
        const float* W  = sWp + mat * 192;
        float*     dst  = (mat == 0) ? sV : (mat == 1) ? sK : sQ;

        const float* xrow = Xs + (rt * 16 + l16) * XPITCH;
        v8f acc = {};
#pragma unroll
        for (int k0 = 0; k0 < 12; k0 += 4) {
            const int ka = k0 + 2 * half;
            v2f a, b;
            a.x = xrow[ka];
            a.y = xrow[ka + 1];
            b.x = W[l16 * XPITCH + ka];
            b.y = W[l16 * XPITCH + ka + 1];
            acc = __builtin_amdgcn_wmma_f32_16x16x4_f32(
                false, a, false, b, (short)0, acc, false, false);
        }
        const int rbase = rt * 16 + half * 8;
#pragma unroll
        for (int r = 0; r < 8; ++r)
            dst[(rbase + r) * QKV_PITCH + l16] = acc[r];
    }
    __syncthreads();

    // ---- Phase 2: per-(n,h,q) attention row on VALU.
    if (tid < nv * 36) {
        const int nl = tid / 36, u = tid % 36;
        const int h = u / 9, q = u % 9;
        const float inv_sqrt_e = 1.0f / 6.0f;     // 1/sqrt(E=36)
        float qr[9];
#pragma unroll
        for (int d = 0; d < 9; ++d)
            qr[d] = sQ[(nl * 36 + q * 4 + h) * QKV_PITCH + d];
        float e[9];
        float m = -1e30f;
#pragma unroll
        for (int l = 0; l < 9; ++l) {
            float s = 0.f;
#pragma unroll
            for (int d = 0; d < 9; ++d)
                s += qr[d] * sK[(nl * 36 + l * 4 + h) * QKV_PITCH + d];
            e[l] = s * inv_sqrt_e;
            m = fmaxf(m, e[l]);
        }
        float sum = 0.f;
#pragma unroll
        for (int l = 0; l < 9; ++l) { e[l] = __expf(e[l] - m); sum += e[l]; }
        const float inv = 1.f / sum;
        float o[9] = {0.f, 0.f, 0.f, 0.f, 0.f, 0.f, 0.f, 0.f, 0.f};
#pragma unroll
        for (int l = 0; l < 9; ++l) {
            const float a = e[l] * inv;
#pragma unroll
            for (int d = 0; d < 9; ++d)
                o[d] += a * sV[(nl * 36 + l * 4 + h) * QKV_PITCH + d];
        }
#pragma unroll
        for (int d = 0; d < 9; ++d)
            sO[(nl * 9 + q) * 36 + h * 9 + d] = o[d];
    }
    __syncthreads();

    // ---- Phase 3: Y(36x36) = O(36x36) @ Wo^T + bo via WMMA, K=36 = 9 exact steps.
    for (int t = wave; t < 9; t += 8) {
        const int rt = t / 3, ct = t % 3;
        const int rloc = rt * 16 + l16;           // 0..47 (36 valid)
        const int jg   = ct * 16 + l16;           // 0..47 (36 valid)
        const bool jv  = jg < 36;
        // Clamped (select) bases: garbage rows/cols only feed D entries we skip.
        const float* arow = sO + (rloc < 36 ? rloc : 0) * 36;
        const float* brow = sWo + (jv ? jg : 0) * 36;
        v8f acc = {};
#pragma unroll
        for (int k0 = 0; k0 < 36; k0 += 4) {
            const int ka = k0 + 2 * half;
            v2f a, b;
            a.x = arow[ka];
            a.y = arow[ka + 1];
            b.x = brow[ka];
            b.y = brow[ka + 1];
            acc = __builtin_amdgcn_wmma_f32_16x16x4_f32(
                false, a, false, b, (short)0, acc, false, false);
        }
        const float bias = sbo[jv ? jg : 0];
#pragma unroll
        for (int r = 0; r < 8; ++r) {
            const int row = rt * 16 + half * 8 + r;
            if (row < 36 && jv) {
                const int nl = row / 9, s = row % 9;
                sY[nl * 324 + jg * 9 + s] = acc[r] + bias;   // == out layout
            }
        }
    }
    __syncthreads();

    // ---- Phase 4: coalesced b128 store of the block's contiguous output region.
    {
        float4* og = (float4*)(out + (size_t)n0 * 324);
        const int nf4 = nv * 81;
        for (int i = tid; i < nf4; i += 256)
            og[i] = ((const float4*)sY)[i];
    }
}

extern "C" void kernel_launch(void* const* d_in, const int* in_sizes, int n_in,
                              void* d_out, int out_size, void* d_ws, size_t ws_size,
                              hipStream_t stream) {
    const float* values = (const float*)d_in[0];
    const float* keys   = (const float*)d_in[1];
    const float* query  = (const float*)d_in[2];
    const float* Wv     = (const float*)d_in[3];
    const float* Wk     = (const float*)d_in[4];
    const float* Wq     = (const float*)d_in[5];
    const float* Wo     = (const float*)d_in[6];
    const float* bo     = (const float*)d_in[7];
    // d_in[8] = mask scalar: with S=9, PAD=8 the band covers everything -> no-op.
    float* out = (float*)d_out;

    const int n_total = in_sizes[0] / 324;        // (N, 36, 9)
    const int blocks  = (n_total + NB - 1) / NB;
    attn_kernel<<<blocks, 256, 0, stream>>>(values, keys, query,
                                            Wv, Wk, Wq, Wo, bo, out, n_total);
}